// ODEFunc_86199993631130
// MI455X (gfx1250) — compile-verified
//
#include <hip/hip_runtime.h>

#define TT 1024
#define BB 512
#define HH 20
#define NBCOL 16          // batch columns per block
#define IH_ROWS 1008      // history rows ever read by a GEMM (k < 16*63)
#define IH_STRIDE 1009    // pad for LDS bank spread
#define NPROD 3           // producer waves per block (waves 1..3)

typedef __attribute__((ext_vector_type(2))) float v2f;
typedef __attribute__((ext_vector_type(8))) float v8f;

// ---------------------------------------------------------------------------
// Phase 1: memory-kernel MLP  me = sigmoid(tanh(tanh(tanh(t*W1+b1)W2+b2)W3+b3)W4+b4)
// ---------------------------------------------------------------------------
__global__ __launch_bounds__(256) void memory_mlp_kernel(
    const float* __restrict__ t,
    const float* __restrict__ w1, const float* __restrict__ b1,
    const float* __restrict__ w2, const float* __restrict__ b2,
    const float* __restrict__ w3, const float* __restrict__ b3,
    const float* __restrict__ w4, const float* __restrict__ b4,
    float* __restrict__ me)
{
    int i = blockIdx.x * blockDim.x + threadIdx.x;
    if (i >= TT) return;
    float tv = t[i];
    float h1[HH], h2[HH];
    #pragma unroll
    for (int j = 0; j < HH; ++j) h1[j] = tanhf(fmaf(tv, w1[j], b1[j]));
    for (int j = 0; j < HH; ++j) {
        float s = b2[j];
        #pragma unroll
        for (int k = 0; k < HH; ++k) s = fmaf(h1[k], w2[k * HH + j], s);
        h2[j] = tanhf(s);
    }
    for (int j = 0; j < HH; ++j) {
        float s = b3[j];
        #pragma unroll
        for (int k = 0; k < HH; ++k) s = fmaf(h2[k], w3[k * HH + j], s);
        h1[j] = tanhf(s);
    }
    float s = b4[0];
    #pragma unroll
    for (int k = 0; k < HH; ++k) s = fmaf(h1[k], w4[k], s);
    me[i] = 1.0f / (1.0f + expf(-s));
}

// ---------------------------------------------------------------------------
// Phase 2: integro-ODE scan, producer/consumer pipelined across 4 waves.
//   wave 0 (consumer): per tile t -- 16-row leftover WMMA chunk, sum producer
//                      partials from LDS, run the 16 serial ODE steps.
//   waves 1..3 (producers): during tile t, compute tile t+1's old-history
//                      GEMM over k < 16t (rows already in LDS), interleaved
//                      by 16-K chunks, into double-buffered LDS partials.
// Old history:  acc[jj][b] = sum_{k<16t} me[TT-(16t+jj)+k] * Ihist[k][b]
// via V_WMMA_F32_16X16X4_F32 (A = me shifts 16x4, B = I_hist 4x16, f32).
// ---------------------------------------------------------------------------
__global__ __launch_bounds__(128) void scan_kernel(
    const float* __restrict__ t,
    const float* __restrict__ y,
    const float* __restrict__ me_g,
    const float* __restrict__ beta_p,
    const float* __restrict__ gamma_p,
    float* __restrict__ out)   // [solution T*B*3 | diff T*B*3]
{
    __shared__ float meS[TT];                      //  4,096 B
    __shared__ float IhistT[NBCOL][IH_STRIDE];     // 64,576 B
    __shared__ float accbuf[2][NPROD][16][16];     //  6,144 B  (total 74,816 B)

    const int tid      = threadIdx.x;
    const int wave     = tid >> 5;         // 0 consumer, 1..3 producers
    const int lane     = tid & 31;
    const int col      = lane & 15;        // batch column / matrix N (and A-row)
    const int laneHalf = lane >> 4;        // fragment half: K +0/+2
    const int gb       = blockIdx.x * NBCOL + col;

    const float beta = beta_p[0];
    const float gmma = gamma_p[0];
    const float dt   = t[0] - t[1];

    for (int i = tid; i < TT; i += 128) meS[i] = me_g[i];
    {   // zero the partial buffers (tile 0 reads zeros)
        float* ab = &accbuf[0][0][0][0];
        for (int i = tid; i < 2 * NPROD * 256; i += 128) ab[i] = 0.f;
    }
    __syncthreads();

    // tile-invariant tail weights: within-tile term uses me[TT - m], m=1..15
    float mtail[16];
    mtail[0] = 0.0f;
    #pragma unroll
    for (int m = 1; m < 16; ++m) mtail[m] = meS[TT - m];

    float S = 0.f, I = 0.f, R = 0.f;
    if (wave == 0 && laneHalf == 0) {
        S = y[gb * 3 + 0];
        I = y[gb * 3 + 1];
        R = y[gb * 3 + 2];
        // solution[0] = y0
        out[(size_t)gb * 3 + 0] = S;
        out[(size_t)gb * 3 + 1] = I;
        out[(size_t)gb * 3 + 2] = R;
        // diff[T-1] = 0
        size_t dz = (size_t)TT * BB * 3 + ((size_t)(TT - 1) * BB + gb) * 3;
        out[dz + 0] = 0.f; out[dz + 1] = 0.f; out[dz + 2] = 0.f;
    }
    __syncthreads();

    float Irec[16];   // consumer: this tile's fresh I values (per column/lane)

    for (int tile = 0; tile < TT / 16; ++tile) {
        const int Kold  = 16 * tile;
        const int bbase = 2 * laneHalf;

        if (wave == 0) {
            // ---- consumer: leftover chunk k in [Kold-16, Kold) for THIS tile
            v8f acc0 = {0.f,0.f,0.f,0.f,0.f,0.f,0.f,0.f};
            v8f acc1 = {0.f,0.f,0.f,0.f,0.f,0.f,0.f,0.f};
            if (tile > 0) {
                const int abase = TT - Kold - col + 2 * laneHalf;
                const int k0 = Kold - 16;
                #pragma unroll
                for (int sub = 0; sub < 16; sub += 8) {
                    const int kk = k0 + sub;
                    v2f a0, b0, a1, b1;
                    a0.x = meS[abase + kk];     a0.y = meS[abase + kk + 1];
                    b0.x = IhistT[col][bbase + kk];
                    b0.y = IhistT[col][bbase + kk + 1];
                    a1.x = meS[abase + kk + 4]; a1.y = meS[abase + kk + 5];
                    b1.x = IhistT[col][bbase + kk + 4];
                    b1.y = IhistT[col][bbase + kk + 5];
                    acc0 = __builtin_amdgcn_wmma_f32_16x16x4_f32(
                        false, a0, false, b0, (short)0, acc0, false, false);
                    acc1 = __builtin_amdgcn_wmma_f32_16x16x4_f32(
                        false, a1, false, b1, (short)0, acc1, false, false);
                }
            }
            // ---- sum producer partials (written during tile-1 into buf tile&1)
            const int bufr = tile & 1;
            v8f acc;
            #pragma unroll
            for (int v = 0; v < 8; ++v) {
                const int row = v + 8 * laneHalf;
                float p = accbuf[bufr][0][row][col]
                        + accbuf[bufr][1][row][col]
                        + accbuf[bufr][2][row][col];
                acc[v] = (acc0[v] + acc1[v]) + p;
            }

            // ---- move accumulator rows to serial lanes (lanes 0..15)
            float accRow[16];
            #pragma unroll
            for (int jj = 0; jj < 8; ++jj)  accRow[jj] = acc[jj];
            #pragma unroll
            for (int jj = 8; jj < 16; ++jj) accRow[jj] = __shfl_down(acc[jj - 8], 16, 32);

            // ---- serial in-tile steps (lanes 0..15, one column each)
            if (laneHalf == 0) {
                #pragma unroll
                for (int jj = 0; jj < 16; ++jj) {
                    const int j = 16 * tile + jj;
                    if (j == 0) {
                        Irec[0] = I;
                        IhistT[col][0] = I;     // I_hist[0] = y0's I
                    } else {
                        // oldest terms first: only the final FMA depends on
                        // the previous step's freshly produced I.
                        float within = 0.f;
                        #pragma unroll
                        for (int m = jj; m >= 1; --m)
                            within = fmaf(Irec[jj - m], mtail[m], within);
                        const float integro = dt * (accRow[jj] + within);
                        const float bSI = beta * S * I;
                        const float dS  = integro - bSI;
                        const float dI  = bSI - gmma * I;
                        const float dR  = gmma * I - integro;
                        S = fmaf(dS, dt, S);
                        I = fmaf(dI, dt, I);
                        R = fmaf(dR, dt, R);
                        Irec[jj] = I;
                        if (j < IH_ROWS) IhistT[col][j] = I;
                        const size_t sb = ((size_t)j * BB + gb) * 3;
                        out[sb + 0] = S; out[sb + 1] = I; out[sb + 2] = R;
                        const size_t db = (size_t)TT * BB * 3
                                        + ((size_t)(j - 1) * BB + gb) * 3;
                        out[db + 0] = dS; out[db + 1] = dI; out[db + 2] = dR;
                    }
                }
            }
        } else {
            // ---- producers: old-history GEMM for tile t+1 over k < 16t
            // (rows < 16t were completed by the end of tile t-1's serial phase)
            const int w = wave - 1;
            v8f acc0 = {0.f,0.f,0.f,0.f,0.f,0.f,0.f,0.f};
            v8f acc1 = {0.f,0.f,0.f,0.f,0.f,0.f,0.f,0.f};
            if (tile < (TT / 16) - 1) {
                const int abaseN = TT - 16 * (tile + 1) - col + 2 * laneHalf;
                for (int k0 = 16 * w; k0 < Kold; k0 += 16 * NPROD) {
                    #pragma unroll
                    for (int sub = 0; sub < 16; sub += 8) {
                        const int kk = k0 + sub;
                        v2f a0, b0, a1, b1;
                        a0.x = meS[abaseN + kk];     a0.y = meS[abaseN + kk + 1];
                        b0.x = IhistT[col][bbase + kk];
                        b0.y = IhistT[col][bbase + kk + 1];
                        a1.x = meS[abaseN + kk + 4]; a1.y = meS[abaseN + kk + 5];
                        b1.x = IhistT[col][bbase + kk + 4];
                        b1.y = IhistT[col][bbase + kk + 5];
                        acc0 = __builtin_amdgcn_wmma_f32_16x16x4_f32(
                            false, a0, false, b0, (short)0, acc0, false, false);
                        acc1 = __builtin_amdgcn_wmma_f32_16x16x4_f32(
                            false, a1, false, b1, (short)0, acc1, false, false);
                    }
                }
            }
            // write partials into the buffer the consumer reads NEXT tile
            const int bufw = (tile + 1) & 1;
            #pragma unroll
            for (int v = 0; v < 8; ++v)
                accbuf[bufw][w][v + 8 * laneHalf][col] = acc0[v] + acc1[v];
        }
        __syncthreads();   // partials + new history rows visible for next tile
    }
}

// ---------------------------------------------------------------------------
extern "C" void kernel_launch(void* const* d_in, const int* in_sizes, int n_in,
                              void* d_out, int out_size, void* d_ws, size_t ws_size,
                              hipStream_t stream) {
    const float* t     = (const float*)d_in[0];
    const float* y     = (const float*)d_in[1];
    const float* g_w1  = (const float*)d_in[2];
    const float* g_b1  = (const float*)d_in[3];
    const float* g_w2  = (const float*)d_in[4];
    const float* g_b2  = (const float*)d_in[5];
    const float* g_w3  = (const float*)d_in[6];
    const float* g_b3  = (const float*)d_in[7];
    const float* g_w4  = (const float*)d_in[8];
    const float* g_b4  = (const float*)d_in[9];
    const float* beta  = (const float*)d_in[10];
    const float* gamma = (const float*)d_in[11];

    float* out = (float*)d_out;
    float* me  = (float*)d_ws;   // 1024 floats of scratch

    memory_mlp_kernel<<<TT / 256, 256, 0, stream>>>(
        t, g_w1, g_b1, g_w2, g_b2, g_w3, g_b3, g_w4, g_b4, me);

    scan_kernel<<<BB / NBCOL, 128, 0, stream>>>(t, y, me, beta, gamma, out);
}